// Window_Attention_73701638800163
// MI455X (gfx1250) — compile-verified
//
#include <hip/hip_runtime.h>
#include <math.h>

typedef float v2f __attribute__((ext_vector_type(2)));
typedef float v8f __attribute__((ext_vector_type(8)));

#define NB   16
#define NC   3
#define NH   512
#define NW   512
#define WS   16
#define HWIN 256     // ws*ws
#define HID  12

__global__ __launch_bounds__(256) void winattn_fused(
    const float* __restrict__ x,      const float* __restrict__ fhigh,
    const float* __restrict__ n1w,    const float* __restrict__ n1b,
    const float* __restrict__ qcw,    const float* __restrict__ qdw,
    const float* __restrict__ kvcw,   const float* __restrict__ kvdw,
    const float* __restrict__ tempp,
    const float* __restrict__ pcw,    const float* __restrict__ pcb,
    const float* __restrict__ n2w,    const float* __restrict__ n2b,
    const float* __restrict__ f1w,    const float* __restrict__ f1b,
    const float* __restrict__ f2w,    const float* __restrict__ f2b,
    float* __restrict__ out)
{
    __shared__ float s_q [NC * HWIN];
    __shared__ float s_kv[NC * HWIN];
    __shared__ float s_ln[HWIN * 4];     // LN2 output (K-padded to 4); reused for MLP out
    __shared__ float s_h [HWIN * 16];    // hidden activations (12 padded to 16)
    __shared__ float s_red[15 * 8];

    const int w   = blockIdx.x;          // window id = b*1024 + wy*32 + wx
    const int b   = w >> 10;
    const int wy  = (w >> 5) & 31;
    const int wx  = w & 31;
    const int tid = threadIdx.x;
    const int i   = tid >> 4;
    const int j   = tid & 15;
    const int lane = tid & 31;
    const int wave = tid >> 5;

    const int  y     = wy * WS + i;
    const int  xg    = wx * WS + j;
    const long pix   = (long)y * NW + xg;
    const long plane = (long)NH * NW;
    const long xbase = (long)b * NC * plane + pix;

    // ---- load x (3 channels), LayerNorm1 over C ----
    float xin[3];
    xin[0] = x[xbase];
    xin[1] = x[xbase + plane];
    xin[2] = x[xbase + 2 * plane];
    {
        float m  = (xin[0] + xin[1] + xin[2]) * (1.f / 3.f);
        float d0 = xin[0] - m, d1 = xin[1] - m, d2 = xin[2] - m;
        float v  = (d0 * d0 + d1 * d1 + d2 * d2) * (1.f / 3.f);
        float rs = rsqrtf(v + 1e-5f);
        float xn0 = d0 * rs * n1w[0] + n1b[0];
        float xn1 = d1 * rs * n1w[1] + n1b[1];
        float xn2 = d2 * rs * n1w[2] + n1b[2];
        // conv1x1 kv -> LDS
        #pragma unroll
        for (int c = 0; c < 3; ++c)
            s_kv[c * HWIN + tid] = kvcw[c*3+0]*xn0 + kvcw[c*3+1]*xn1 + kvcw[c*3+2]*xn2;
    }
    {
        const long fb = (long)w * (NC * HWIN);
        float f0 = fhigh[fb + tid];
        float f1 = fhigh[fb + HWIN + tid];
        float f2 = fhigh[fb + 2 * HWIN + tid];
        #pragma unroll
        for (int c = 0; c < 3; ++c)
            s_q[c * HWIN + tid] = qcw[c*3+0]*f0 + qcw[c*3+1]*f1 + qcw[c*3+2]*f2;
    }
    __syncthreads();

    // ---- depthwise 3x3 (zero pad at window edges) ----
    float qd[3], kd[3];
    #pragma unroll
    for (int c = 0; c < 3; ++c) {
        float aq = 0.f, ak = 0.f;
        #pragma unroll
        for (int di = 0; di < 3; ++di) {
            int ii = i + di - 1;
            if (ii < 0 || ii > 15) continue;
            #pragma unroll
            for (int dj = 0; dj < 3; ++dj) {
                int jj = j + dj - 1;
                if (jj < 0 || jj > 15) continue;
                int idx = c * HWIN + ii * 16 + jj;
                aq += qdw [c*9 + di*3 + dj] * s_q [idx];
                ak += kvdw[c*9 + di*3 + dj] * s_kv[idx];
            }
        }
        qd[c] = aq; kd[c] = ak;
    }

    // ---- 15 block reductions over 256 pixels ----
    float vals[15];
    #pragma unroll
    for (int c = 0; c < 3; ++c) { vals[c] = qd[c]*qd[c]; vals[3+c] = kd[c]*kd[c]; }
    #pragma unroll
    for (int c = 0; c < 3; ++c)
        #pragma unroll
        for (int d = 0; d < 3; ++d) vals[6 + c*3 + d] = qd[c] * kd[d];

    #pragma unroll
    for (int r = 0; r < 15; ++r) {
        float v = vals[r];
        #pragma unroll
        for (int off = 16; off > 0; off >>= 1) v += __shfl_down(v, off);
        if (lane == 0) s_red[r * 8 + wave] = v;
    }
    __syncthreads();

    float sums[15];
    #pragma unroll
    for (int r = 0; r < 15; ++r) {
        float s = 0.f;
        #pragma unroll
        for (int k = 0; k < 8; ++k) s += s_red[r * 8 + k];
        sums[r] = s;
    }

    // ---- 3x3 attention (redundant per thread, trivial) ----
    const float temp = tempp[0];
    float nq[3], nk[3];
    #pragma unroll
    for (int c = 0; c < 3; ++c) {
        nq[c] = fmaxf(sqrtf(sums[c]),     1e-12f);
        nk[c] = fmaxf(sqrtf(sums[3 + c]), 1e-12f);
    }
    float attn[3][3];
    #pragma unroll
    for (int c = 0; c < 3; ++c) {
        float l[3];
        #pragma unroll
        for (int d = 0; d < 3; ++d) l[d] = sums[6 + c*3 + d] * temp / (nq[c] * nk[d]);
        float mx = fmaxf(l[0], fmaxf(l[1], l[2]));
        float e0 = __expf(l[0]-mx), e1 = __expf(l[1]-mx), e2 = __expf(l[2]-mx);
        float inv = 1.f / (e0 + e1 + e2);
        attn[c][0] = e0*inv; attn[c][1] = e1*inv; attn[c][2] = e2*inv;
    }

    // x2 = attn @ v  (v = kd, unnormalized), then proj conv1x1 + bias + residual
    float x2c[3], xs2[3];
    #pragma unroll
    for (int c = 0; c < 3; ++c)
        x2c[c] = attn[c][0]*kd[0] + attn[c][1]*kd[1] + attn[c][2]*kd[2];
    #pragma unroll
    for (int c = 0; c < 3; ++c)
        xs2[c] = xin[c] + pcw[c*3+0]*x2c[0] + pcw[c*3+1]*x2c[1] + pcw[c*3+2]*x2c[2] + pcb[c];

    // ---- LayerNorm2, stage into WMMA A layout (K padded 3->4) ----
    {
        float m  = (xs2[0] + xs2[1] + xs2[2]) * (1.f / 3.f);
        float d0 = xs2[0]-m, d1 = xs2[1]-m, d2 = xs2[2]-m;
        float v  = (d0*d0 + d1*d1 + d2*d2) * (1.f / 3.f);
        float rs = rsqrtf(v + 1e-5f);
        s_ln[tid*4 + 0] = d0 * rs * n2w[0] + n2b[0];
        s_ln[tid*4 + 1] = d1 * rs * n2w[1] + n2b[1];
        s_ln[tid*4 + 2] = d2 * rs * n2w[2] + n2b[2];
        s_ln[tid*4 + 3] = 0.f;
    }
    __syncthreads();

    // ---- MLP via V_WMMA_F32_16X16X4_F32 ----
    // A layout (16x4 f32): lanes 0-15 hold K={0,1}, lanes 16-31 hold K={2,3}; M = lane&15
    // D layout (16x16 f32): VGPR r -> M = r (lanes 0-15) / r+8 (lanes 16-31); N = lane&15
    const int half = lane >> 4;
    const int row  = lane & 15;
    const int koff = half * 2;

    // fc1: [16 pixels x 3] @ [3 x 12]  (B[k][n] = fc1_w[n][k], zero-padded)
    v2f b1;
    b1.x = (koff     < 3 && row < HID) ? f1w[row*3 + koff    ] : 0.f;
    b1.y = (koff + 1 < 3 && row < HID) ? f1w[row*3 + koff + 1] : 0.f;
    const float bias1 = (row < HID) ? f1b[row] : 0.f;

    #pragma unroll
    for (int t = 0; t < 2; ++t) {
        const int pbase = wave * 32 + t * 16;
        const int p = pbase + row;
        v2f av; av.x = s_ln[p*4 + koff]; av.y = s_ln[p*4 + koff + 1];
        v8f acc = {};
        acc = __builtin_amdgcn_wmma_f32_16x16x4_f32(false, av, false, b1,
                                                    (short)0, acc, false, false);
        #pragma unroll
        for (int r = 0; r < 8; ++r) {
            float hv = acc[r] + bias1;
            hv = 0.5f * hv * (1.f + erff(hv * 0.70710678118654752f));   // exact GELU
            s_h[(pbase + r + half*8) * 16 + row] = hv;                  // j>=12 columns stay 0
        }
    }
    __syncthreads();

    // fc2: [16 pixels x 12] @ [12 x 3], K split into 3 chunks of 4, chained accumulators
    #pragma unroll
    for (int t = 0; t < 2; ++t) {
        const int pbase = wave * 32 + t * 16;
        const int p = pbase + row;
        v8f acc = {};
        #pragma unroll
        for (int kc = 0; kc < 3; ++kc) {
            v2f av; av.x = s_h[p*16 + kc*4 + koff]; av.y = s_h[p*16 + kc*4 + koff + 1];
            const int kg = kc * 4 + koff;
            v2f bv;
            bv.x = (kg     < HID && row < 3) ? f2w[row*HID + kg    ] : 0.f;
            bv.y = (kg + 1 < HID && row < 3) ? f2w[row*HID + kg + 1] : 0.f;
            acc = __builtin_amdgcn_wmma_f32_16x16x4_f32(false, av, false, bv,
                                                        (short)0, acc, false, false);
        }
        if (row < 3) {
            #pragma unroll
            for (int r = 0; r < 8; ++r)
                s_ln[(pbase + r + half*8) * 4 + row] = acc[r];   // reuse s_ln as MLP-out buf
        }
    }
    __syncthreads();

    // ---- residual + coalesced channel-plane stores ----
    #pragma unroll
    for (int c = 0; c < 3; ++c)
        out[(long)(b * NC + c) * plane + pix] = xs2[c] + s_ln[tid*4 + c] + f2b[c];
}

extern "C" void kernel_launch(void* const* d_in, const int* in_sizes, int n_in,
                              void* d_out, int out_size, void* d_ws, size_t ws_size,
                              hipStream_t stream) {
    (void)in_sizes; (void)n_in; (void)out_size; (void)d_ws; (void)ws_size;
    // setup_inputs order: 0:H 1:W 2:x 3:fhigh 4:norm1_w 5:norm1_b 6:q_C_w 7:q_dw_w
    // 8:kv_C_w 9:kv_dw_w 10:temperature 11:proj_C_w 12:proj_C_b 13:norm2_w 14:norm2_b
    // 15:fc1_w 16:fc1_b 17:fc2_w 18:fc2_b
    const float* x     = (const float*)d_in[2];
    const float* fh    = (const float*)d_in[3];
    const float* n1w   = (const float*)d_in[4];
    const float* n1b   = (const float*)d_in[5];
    const float* qcw   = (const float*)d_in[6];
    const float* qdw   = (const float*)d_in[7];
    const float* kvcw  = (const float*)d_in[8];
    const float* kvdw  = (const float*)d_in[9];
    const float* tempp = (const float*)d_in[10];
    const float* pcw   = (const float*)d_in[11];
    const float* pcb   = (const float*)d_in[12];
    const float* n2w   = (const float*)d_in[13];
    const float* n2b   = (const float*)d_in[14];
    const float* f1w   = (const float*)d_in[15];
    const float* f1b   = (const float*)d_in[16];
    const float* f2w   = (const float*)d_in[17];
    const float* f2b   = (const float*)d_in[18];
    float* out = (float*)d_out;

    const int nWindows = NB * (NH / WS) * (NW / WS);   // 16384
    winattn_fused<<<nWindows, 256, 0, stream>>>(
        x, fh, n1w, n1b, qcw, qdw, kvcw, kvdw, tempp,
        pcw, pcb, n2w, n2b, f1w, f1b, f2w, f2b, out);
}